// BaseAggregator_31439160607279
// MI455X (gfx1250) — compile-verified
//
#include <hip/hip_runtime.h>
#include <hip/hip_bf16.h>

typedef __attribute__((ext_vector_type(2))) float v2f;
typedef __attribute__((ext_vector_type(8))) float v8f;

#define A_N     32
#define V_N     32
#define C_FULL  512
#define K_HEADS 2
#define NC      256     // channels per head
#define T_N     200     // audio time (f=1)
#define HW_N    196     // 14*14 image positions
#define KC      32      // K-chunk staged in LDS
#define PADW    224     // padded width: 14 tiles of 16 (N side uses 14th as zero pad)
#define NWAVES  7       // each wave owns TWO 16-wide t tiles
#define BLOCK   (NWAVES * 32)
#define MT_N    13      // hw tiles of 16 (196 -> 208)

__global__ __launch_bounds__(BLOCK, 1)
void sim_agg_kernel(const float* __restrict__ audio,   // (32, 512, 1, 200)
                    const float* __restrict__ image,   // (32, 512, 14, 14)
                    const float* __restrict__ acls,    // (32, 512)
                    const float* __restrict__ icls,    // (32, 512)
                    const int*   __restrict__ mask,    // (32, 200)
                    float*       __restrict__ out)     // (32, 32)
{
    // LDS chunks interleaved: element (c, p) at [(c>>2)*PADW + p]*4 + (c&3)
    // -> each lane's (K,K+1) fragment pair is one contiguous ds_load_b64.
    __shared__ float ls_im[(KC / 4) * PADW * 4];   // 7168 floats
    __shared__ float ls_au[(KC / 4) * PADW * 4];   // 7168 floats
    __shared__ float s_cls[K_HEADS];
    __shared__ float s_red[NWAVES];

    const int a      = blockIdx.x >> 5;
    const int v      = blockIdx.x & 31;
    const int tid    = threadIdx.x;
    const int wid    = tid >> 5;            // 0..6
    const int lane   = tid & 31;
    const int laneLo = lane & 15;
    const int hi2    = (lane >> 4) << 1;    // K offset {0,2} per lane half
    const int n0     = 2 * wid;             // this wave's two t tiles
    const int n1     = 2 * wid + 1;         // tile 13 (wave 6) is zero padding

    // ---- per-(a,v,k) CLS similarity (tiny dot products) ----
    if (wid < K_HEADS) {
        const float* ap = acls + a * C_FULL + wid * NC;
        const float* ip = icls + v * C_FULL + wid * NC;
        float p = 0.f;
        for (int j = lane; j < NC; j += 32) p += ap[j] * ip[j];
        for (int off = 16; off >= 1; off >>= 1) p += __shfl_xor(p, off, 32);
        if (lane == 0) s_cls[wid] = p;
    }
    __syncthreads();

    float wave_num = 0.f;

    for (int k = 0; k < K_HEADS; ++k) {
        v8f accA[MT_N];   // N-tile n0
        v8f accB[MT_N];   // N-tile n1
#pragma unroll
        for (int mt = 0; mt < MT_N; ++mt) {
            accA[mt] = (v8f){0.f,0.f,0.f,0.f,0.f,0.f,0.f,0.f};
            accB[mt] = (v8f){0.f,0.f,0.f,0.f,0.f,0.f,0.f,0.f};
        }

        for (int cb = 0; cb < NC; cb += KC) {
            // ---- stage K-chunk of Im (196 wide) and Au (200 wide) into LDS ----
            const float* imB = image + ((size_t)v * C_FULL + k * NC + cb) * HW_N;
            const float* auB = audio + ((size_t)a * C_FULL + k * NC + cb) * T_N;
            for (int i = tid; i < KC * PADW; i += BLOCK) {  // 7168 = 224*32, uniform
                int c = i / PADW;
                int p = i - c * PADW;
                ls_im[((c >> 2) * PADW + p) * 4 + (c & 3)] = (p < HW_N) ? imB[c * HW_N + p] : 0.f;
                ls_au[((c >> 2) * PADW + p) * 4 + (c & 3)] = (p < T_N)  ? auB[c * T_N  + p] : 0.f;
            }
            __syncthreads();

            // ---- K steps of 4: each A-frag feeds TWO wmmas (halved DS traffic) ----
            for (int c4 = 0; c4 < KC / 4; ++c4) {
                const float* bp0 = &ls_au[(c4 * PADW + n0 * 16 + laneLo) * 4 + hi2];
                const float* bp1 = &ls_au[(c4 * PADW + n1 * 16 + laneLo) * 4 + hi2];
                v2f b0; b0.x = bp0[0]; b0.y = bp0[1];
                v2f b1; b1.x = bp1[0]; b1.y = bp1[1];
                v2f af[MT_N];
#pragma unroll
                for (int mt = 0; mt < MT_N; ++mt) {   // batched: loads issue as a clause
                    const float* apf = &ls_im[(c4 * PADW + mt * 16 + laneLo) * 4 + hi2];
                    af[mt].x = apf[0]; af[mt].y = apf[1];
                }
#pragma unroll
                for (int mt = 0; mt < MT_N; ++mt) {
                    accA[mt] = __builtin_amdgcn_wmma_f32_16x16x4_f32(
                        false, af[mt], false, b0, (short)0, accA[mt], false, false);
                    accB[mt] = __builtin_amdgcn_wmma_f32_16x16x4_f32(
                        false, af[mt], false, b1, (short)0, accB[mt], false, false);
                }
            }
            __syncthreads();
        }

        // ---- column max over hw (C layout: VGPR r -> M=r | M=r+8 per lane half) ----
        float cmax0 = -__builtin_inff();
        float cmax1 = -__builtin_inff();
        const int hwAdd = (lane >> 4) << 3;
#pragma unroll
        for (int mt = 0; mt < MT_N; ++mt) {
#pragma unroll
            for (int r = 0; r < 8; ++r) {
                int hwRow = mt * 16 + r + hwAdd;
                if (hwRow < HW_N) {                     // folds for mt < 12
                    cmax0 = fmaxf(cmax0, accA[mt][r]);
                    cmax1 = fmaxf(cmax1, accB[mt][r]);
                }
            }
        }
        cmax0 = fmaxf(cmax0, __shfl_xor(cmax0, 16, 32));   // merge lane halves
        cmax1 = fmaxf(cmax1, __shfl_xor(cmax1, 16, 32));

        // ---- relu(max + cls) * mask over this wave's 32 t columns ----
        float contrib = 0.f;
        if (lane < 16) {
            int t0 = n0 * 16 + laneLo;
            int t1 = n1 * 16 + laneLo;
            if (t0 < T_N) contrib += fmaxf(cmax0 + s_cls[k], 0.f) * (float)mask[a * T_N + t0];
            if (t1 < T_N) contrib += fmaxf(cmax1 + s_cls[k], 0.f) * (float)mask[a * T_N + t1];
        }
        for (int off = 16; off >= 1; off >>= 1) contrib += __shfl_xor(contrib, off, 32);
        wave_num += contrib;
    }

    // ---- cross-wave reduction, divide by masked denominator ----
    if (lane == 0) s_red[wid] = wave_num;
    __syncthreads();
    if (tid == 0) {
        float tot = 0.f;
        for (int wq = 0; wq < NWAVES; ++wq) tot += s_red[wq];
        float den = 0.f;
        const int* mrow = mask + a * T_N;
        for (int tt = 0; tt < T_N; ++tt) den += (float)mrow[tt];
        out[blockIdx.x] = tot / den;   // out[a*32 + v]
    }
}

extern "C" void kernel_launch(void* const* d_in, const int* in_sizes, int n_in,
                              void* d_out, int out_size, void* d_ws, size_t ws_size,
                              hipStream_t stream) {
    const float* audio = (const float*)d_in[0];
    const float* image = (const float*)d_in[1];
    const float* acls  = (const float*)d_in[2];
    const float* icls  = (const float*)d_in[3];
    const int*   mask  = (const int*)d_in[4];
    // d_in[5] = agg_heads (always 1 for this harness: sum over heads -> (32,32))
    float* outp = (float*)d_out;
    sim_agg_kernel<<<A_N * V_N, BLOCK, 0, stream>>>(audio, image, acls, icls, mask, outp);
}